// ModelNew_17411797418181
// MI455X (gfx1250) — compile-verified
//
#include <hip/hip_runtime.h>
#include <stdint.h>

// Problem constants (from reference): T=32768, D=2048, W=4, S=3, NSTATES=64, B=8
#define T_TOK 32768
#define D_CH  2048
#define W_K   4
#define S_ST  3
#define NST   64
#define BATCH 8

// Tiling for the main kernel
#define TB 16                    // tokens per block
#define DB 256                   // channels (floats) per block tile row
#define ROWS (TB + W_K - 1)      // 19 rows incl. 3-token halo
#define NTHREADS 256             // 8 wave32 waves

__global__ __launch_bounds__(NTHREADS)
void conv1d_varlen_out_kernel(const float* __restrict__ x,
                              const float* __restrict__ weight,
                              const float* __restrict__ conv_states,
                              const int*   __restrict__ qsl,
                              const int*   __restrict__ ci,
                              const int*   __restrict__ ism,
                              const int*   __restrict__ resconn,
                              float*       __restrict__ out)
{
    __shared__ float tile[ROWS * DB];      // 19 KB x-tile staged via async DMA
    __shared__ int   s_qsl[BATCH + 1];
    __shared__ int   s_ci[BATCH];
    __shared__ int   s_ism[BATCH];

    const int tid = threadIdx.x;
    const int d0  = blockIdx.x * DB;       // channel tile base
    const int t0  = blockIdx.y * TB;       // token tile base

    if (tid < BATCH + 1) s_qsl[tid] = qsl[tid];
    if (tid < BATCH)     { s_ci[tid] = ci[tid]; s_ism[tid] = ism[tid]; }

    // ---- async stage x[(t0-3)..(t0+TB-1), d0..d0+DB) into LDS (b128 chunks) ----
    const unsigned lds_base = (unsigned)(uintptr_t)(&tile[0]);
    const int nchunk = ROWS * DB / 4;      // 1216 x 16B transfers
    for (int f = tid; f < nchunk; f += NTHREADS) {
        const int row = f >> 6;            // DB/4 = 64 chunks per row
        const int c4  = (f & 63) << 2;     // float index within row
        int gr = t0 - (W_K - 1) + row;
        if (gr < 0) gr = 0;                // clamped rows are never consumed (within==false)
        const float* gp = x + (size_t)gr * D_CH + d0 + c4;
        const unsigned lds_addr = lds_base + (unsigned)((row * DB + c4) * 4);
        asm volatile("global_load_async_to_lds_b128 %0, %1, off"
                     :: "v"(lds_addr), "v"(gp)
                     : "memory");
    }
    asm volatile("s_wait_asynccnt 0" ::: "memory");  // my waves' ASYNCcnt drained
    __syncthreads();                                  // all waves' tile data visible

    const float rc = (resconn[0] != 0) ? 1.0f : 0.0f;

    const int cid  = tid & 63;            // channel-quad within tile
    const int trow = tid >> 6;            // 0..3
    const int dg   = d0 + (cid << 2);     // global channel base (4 channels/thread)

    // per-channel filter taps: weight row d = 4 contiguous floats -> one float4 each
    const float4 w0 = ((const float4*)weight)[dg + 0];
    const float4 w1 = ((const float4*)weight)[dg + 1];
    const float4 w2 = ((const float4*)weight)[dg + 2];
    const float4 w3 = ((const float4*)weight)[dg + 3];

    #pragma unroll
    for (int k = 0; k < TB / 4; ++k) {
        const int tt = trow + (k << 2);   // 0..15 within tile
        const int t  = t0 + tt;

        // seg = searchsorted(qsl, t, 'right') - 1  == count of qsl[1..B] <= t
        int seg = 0;
        #pragma unroll
        for (int i = 1; i <= BATCH; ++i) seg += (s_qsl[i] <= t) ? 1 : 0;
        const int   start = s_qsl[seg];
        const int   cib   = s_ci[seg];
        const float hi    = (s_ism[seg] != 0) ? 1.0f : 0.0f;

        const float4 xc = *(const float4*)(&tile[(tt + W_K - 1) * DB + (cid << 2)]);

        float4 acc;   // tap j=0 uses weight col 3; residual folded in
        acc.x = xc.x * (w0.w + rc);
        acc.y = xc.y * (w1.w + rc);
        acc.z = xc.z * (w2.w + rc);
        acc.w = xc.w * (w3.w + rc);

        #pragma unroll
        for (int j = 1; j < W_K; ++j) {
            float4 src;
            if (t - j >= start) {
                src = *(const float4*)(&tile[(tt + W_K - 1 - j) * DB + (cid << 2)]);
            } else {
                int sp = S_ST + (t - j) - start;   // in [0, S-1] here
                if (sp < 0) sp = 0;
                if (sp > S_ST - 1) sp = S_ST - 1;
                const float4 v = *(const float4*)(conv_states +
                                   ((size_t)cib * S_ST + sp) * D_CH + dg);
                src.x = v.x * hi; src.y = v.y * hi; src.z = v.z * hi; src.w = v.w * hi;
            }
            // weight column (W-1-j); j-loop is fully unrolled so selects are static
            const float wc0 = (j == 1) ? w0.z : (j == 2) ? w0.y : w0.x;
            const float wc1 = (j == 1) ? w1.z : (j == 2) ? w1.y : w1.x;
            const float wc2 = (j == 1) ? w2.z : (j == 2) ? w2.y : w2.x;
            const float wc3 = (j == 1) ? w3.z : (j == 2) ? w3.y : w3.x;
            acc.x += src.x * wc0;
            acc.y += src.y * wc1;
            acc.z += src.z * wc2;
            acc.w += src.w * wc3;
        }
        *(float4*)(out + (size_t)t * D_CH + dg) = acc;
    }
}

__global__ __launch_bounds__(256)
void conv1d_varlen_state_kernel(const float* __restrict__ x,
                                const float* __restrict__ conv_states,
                                const int*   __restrict__ qsl,
                                const int*   __restrict__ ci,
                                const int*   __restrict__ ism,
                                float*       __restrict__ out_states)
{
    const int n = blockIdx.x;             // state row 0..NST-1
    int b = -1;
    #pragma unroll
    for (int i = 0; i < BATCH; ++i) {
        if (b < 0 && ci[i] == n) b = i;   // first match (ci is unique here)
    }

    const int    tid  = threadIdx.x;
    const size_t base = (size_t)n * S_ST * D_CH;
    const int    nq4  = S_ST * D_CH / 4;  // 1536 float4 per state row

    if (b < 0) {
        // pass-through: conv_states_out row == conv_states row
        for (int f = tid; f < nq4; f += 256)
            ((float4*)(out_states + base))[f] = ((const float4*)(conv_states + base))[f];
        return;
    }

    const int   start = qsl[b];
    const int   end   = qsl[b + 1];
    const float hi    = (ism[b] != 0) ? 1.0f : 0.0f;

    for (int f = tid; f < nq4; f += 256) {
        const int s   = f / (D_CH / 4);
        const int d4  = (f % (D_CH / 4)) << 2;
        const int idx = end - S_ST + s;
        float4 v;
        if (idx >= start) {
            v = *(const float4*)(x + (size_t)idx * D_CH + d4);
        } else {
            int sp = S_ST - start + idx;
            if (sp < 0) sp = 0;
            if (sp > S_ST - 1) sp = S_ST - 1;
            const float4 u = *(const float4*)(conv_states + base + (size_t)sp * D_CH + d4);
            v.x = u.x * hi; v.y = u.y * hi; v.z = u.z * hi; v.w = u.w * hi;
        }
        *(float4*)(out_states + base + (size_t)s * D_CH + d4) = v;
    }
}

extern "C" void kernel_launch(void* const* d_in, const int* in_sizes, int n_in,
                              void* d_out, int out_size, void* d_ws, size_t ws_size,
                              hipStream_t stream) {
    (void)in_sizes; (void)n_in; (void)out_size; (void)d_ws; (void)ws_size;
    const float* x   = (const float*)d_in[0];   // (T, D) f32
    const float* w   = (const float*)d_in[1];   // (D, W) f32
    const float* cs  = (const float*)d_in[2];   // (NSTATES, S, D) f32
    const int*   qsl = (const int*)d_in[3];     // (B+1,) i32
    const int*   ci  = (const int*)d_in[4];     // (B,) i32
    const int*   ism = (const int*)d_in[5];     // (B,) i32
    /* d_in[6] = pad_slot_id (unused by reference math) */
    const int*   rc  = (const int*)d_in[7];     // residual_connection scalar

    float* out        = (float*)d_out;                      // (T, D)
    float* out_states = out + (size_t)T_TOK * D_CH;         // (NSTATES, S, D)

    dim3 gridA(D_CH / DB, T_TOK / TB);                      // 8 x 2048 blocks
    conv1d_varlen_out_kernel<<<gridA, NTHREADS, 0, stream>>>(x, w, cs, qsl, ci, ism, rc, out);
    conv1d_varlen_state_kernel<<<NST, 256, 0, stream>>>(x, cs, qsl, ci, ism, out_states);
}